// DotAtt_40097814675537
// MI455X (gfx1250) — compile-verified
//
#include <hip/hip_runtime.h>
#include <hip/hip_bf16.h>

// X = softmax(mask(Q (K^T V)/sqrt(d)))  -- reassociated: no nonlinearity
// between the two GEMMs in the reference, so we compute M = K^T V once
// (512x512 per batch, 8 MB total in d_ws) and then Q*M fused with softmax.

typedef __attribute__((ext_vector_type(2))) float v2f;
typedef __attribute__((ext_vector_type(8))) float v8f;

#define B_     8
#define SQ     2048
#define SK     2048
#define D_     512
#define DV_    512
#define NEG_FILL  (-1000000.0f)
#define INV_SQRT_D 0.04419417382415922f   // 1/sqrt(512)
#define SM_STRIDE 520                     // 512 + 8 pad: kills LDS bank aliasing

// ---------------------------------------------------------------------------
// Kernel 1: M[b] = (K[b]^T @ V[b]) * (1/sqrt(d))
// Grid: (DV/64, D/128, B), block = 256 (8 waves). Each wave: 16x64 strip.
// WMMA f32 16x16x4: A(16x4)=K^T tile, B(4x16)=V tile, reduce over Sk.
// ---------------------------------------------------------------------------
__global__ void __launch_bounds__(256) kt_v_gemm(const float* __restrict__ K,
                                                 const float* __restrict__ V,
                                                 float* __restrict__ M) {
  const int lane = threadIdx.x & 31;
  const int wave = threadIdx.x >> 5;
  const int m    = lane & 15;
  const int kk0  = (lane >> 4) << 1;               // 0 (lanes 0-15) or 2 (16-31)
  const int b    = blockIdx.z;
  const int p0   = blockIdx.y * 128 + wave * 16;   // output row block (d dim)
  const int q0   = blockIdx.x * 64;                // output col block (dv dim)

  const float* __restrict__ Kb = K + (size_t)b * SK * D_;
  const float* __restrict__ Vb = V + (size_t)b * SK * DV_;
  float*       __restrict__ Mb = M + (size_t)b * D_ * DV_;

  v8f acc0 = {}, acc1 = {}, acc2 = {}, acc3 = {};
  for (int k = 0; k < SK; k += 4) {
    // A[m,kk] = K[b, k+kk, p0+m]   (16x4, M x Kred)
    v2f a;
    a.x = Kb[(size_t)(k + kk0    ) * D_ + p0 + m];
    a.y = Kb[(size_t)(k + kk0 + 1) * D_ + p0 + m];
    // B[kk,n] = V[b, k+kk, q0+n]   (4x16, Kred x N), 4 N-tiles
    const float* v0 = Vb + (size_t)(k + kk0    ) * DV_ + q0 + m;
    const float* v1 = Vb + (size_t)(k + kk0 + 1) * DV_ + q0 + m;
    v2f b0; b0.x = v0[0];  b0.y = v1[0];
    v2f b1; b1.x = v0[16]; b1.y = v1[16];
    v2f b2; b2.x = v0[32]; b2.y = v1[32];
    v2f b3; b3.x = v0[48]; b3.y = v1[48];
    acc0 = __builtin_amdgcn_wmma_f32_16x16x4_f32(false, a, false, b0, (short)0, acc0, false, false);
    acc1 = __builtin_amdgcn_wmma_f32_16x16x4_f32(false, a, false, b1, (short)0, acc1, false, false);
    acc2 = __builtin_amdgcn_wmma_f32_16x16x4_f32(false, a, false, b2, (short)0, acc2, false, false);
    acc3 = __builtin_amdgcn_wmma_f32_16x16x4_f32(false, a, false, b3, (short)0, acc3, false, false);
  }
  // C/D layout: vgpr j -> row j (lanes 0-15) or j+8 (lanes 16-31), col = n0+m
  const int rb = p0 + ((lane >> 4) << 3);
  #pragma unroll
  for (int j = 0; j < 8; ++j) {
    float* dst = Mb + (size_t)(rb + j) * DV_ + q0 + m;
    dst[0]  = acc0[j] * INV_SQRT_D;
    dst[16] = acc1[j] * INV_SQRT_D;
    dst[32] = acc2[j] * INV_SQRT_D;
    dst[48] = acc3[j] * INV_SQRT_D;
  }
}

// ---------------------------------------------------------------------------
// Kernel 2: X = Q @ M, then mask (col > valid_len[row] -> -1e6) + row softmax.
// Grid: (SQ/16, B), block = 256. Wave w computes cols [64w,64w+64) of a
// 16-row tile; tile staged in LDS; then wave w softmaxes rows 2w, 2w+1.
// ---------------------------------------------------------------------------
__global__ void __launch_bounds__(256) qm_softmax(const float* __restrict__ Q,
                                                  const float* __restrict__ M,
                                                  const int* __restrict__ vlen,
                                                  float* __restrict__ out) {
  __shared__ float tile[16 * SM_STRIDE];
  const int lane = threadIdx.x & 31;
  const int wave = threadIdx.x >> 5;
  const int m    = lane & 15;
  const int kk0  = (lane >> 4) << 1;
  const int b    = blockIdx.y;
  const int r0   = blockIdx.x * 16;     // query-row tile
  const int n0   = wave * 64;           // this wave's dv columns

  const float* __restrict__ Qb = Q + (size_t)b * SQ * D_;
  const float* __restrict__ Mb = M + (size_t)b * D_ * DV_;

  v8f acc0 = {}, acc1 = {}, acc2 = {}, acc3 = {};
  const float* qrow = Qb + (size_t)(r0 + m) * D_;   // A tile row for this lane
  for (int p = 0; p < D_; p += 4) {
    v2f a = *(const v2f*)(qrow + p + kk0);          // contiguous b64 load
    const float* mp0 = Mb + (size_t)(p + kk0    ) * DV_ + n0 + m;
    const float* mp1 = Mb + (size_t)(p + kk0 + 1) * DV_ + n0 + m;
    v2f b0; b0.x = mp0[0];  b0.y = mp1[0];
    v2f b1; b1.x = mp0[16]; b1.y = mp1[16];
    v2f b2; b2.x = mp0[32]; b2.y = mp1[32];
    v2f b3; b3.x = mp0[48]; b3.y = mp1[48];
    acc0 = __builtin_amdgcn_wmma_f32_16x16x4_f32(false, a, false, b0, (short)0, acc0, false, false);
    acc1 = __builtin_amdgcn_wmma_f32_16x16x4_f32(false, a, false, b1, (short)0, acc1, false, false);
    acc2 = __builtin_amdgcn_wmma_f32_16x16x4_f32(false, a, false, b2, (short)0, acc2, false, false);
    acc3 = __builtin_amdgcn_wmma_f32_16x16x4_f32(false, a, false, b3, (short)0, acc3, false, false);
  }

  const int rlo = (lane >> 4) << 3;
  #pragma unroll
  for (int j = 0; j < 8; ++j) {
    float* dst = &tile[(rlo + j) * SM_STRIDE + n0 + m];
    dst[0]  = acc0[j];
    dst[16] = acc1[j];
    dst[32] = acc2[j];
    dst[48] = acc3[j];
  }
  __syncthreads();

  // Softmax over dv=512: wave handles rows 2w and 2w+1; lane owns 16 columns.
  #pragma unroll
  for (int rr = 0; rr < 2; ++rr) {
    const int r = wave * 2 + rr;
    const int s = r0 + r;
    const int vl = vlen[s];
    float x[16];
    float mx = -3.402823466e38f;
    #pragma unroll
    for (int kq = 0; kq < 16; ++kq) {
      const int col = lane + (kq << 5);
      float v = tile[r * SM_STRIDE + col];
      if (col > vl) v = NEG_FILL;
      x[kq] = v;
      mx = fmaxf(mx, v);
    }
    #pragma unroll
    for (int off = 16; off >= 1; off >>= 1)
      mx = fmaxf(mx, __shfl_xor(mx, off, 32));
    float sum = 0.0f;
    #pragma unroll
    for (int kq = 0; kq < 16; ++kq) {
      x[kq] = expf(x[kq] - mx);     // masked entries: expf(~-1e6) == 0.0f
      sum += x[kq];
    }
    #pragma unroll
    for (int off = 16; off >= 1; off >>= 1)
      sum += __shfl_xor(sum, off, 32);
    const float inv = 1.0f / sum;
    float* orow = out + ((size_t)b * SQ + s) * DV_;
    #pragma unroll
    for (int kq = 0; kq < 16; ++kq)
      orow[lane + (kq << 5)] = x[kq] * inv;
  }
}

// ---------------------------------------------------------------------------
extern "C" void kernel_launch(void* const* d_in, const int* in_sizes, int n_in,
                              void* d_out, int out_size, void* d_ws, size_t ws_size,
                              hipStream_t stream) {
  // setup_inputs() order: K, V, Q, valid_len
  const float* Kp   = (const float*)d_in[0];
  const float* Vp   = (const float*)d_in[1];
  const float* Qp   = (const float*)d_in[2];
  const int*   vlen = (const int*)d_in[3];
  float*       outp = (float*)d_out;
  float*       Mws  = (float*)d_ws;   // needs B*D*DV*4 = 8 MB

  dim3 g1(DV_ / 64, D_ / 128, B_);    // (8, 4, 8)
  kt_v_gemm<<<g1, 256, 0, stream>>>(Kp, Vp, Mws);

  dim3 g2(SQ / 16, B_);               // (128, 8)
  qm_softmax<<<g2, 256, 0, stream>>>(Qp, Mws, vlen, outp);
}